// DetectNet_10153302688211
// MI455X (gfx1250) — compile-verified
//
#include <hip/hip_runtime.h>
#include <hip/hip_bf16.h>
#include <stdint.h>

typedef float v2f __attribute__((ext_vector_type(2)));
typedef float v8f __attribute__((ext_vector_type(8)));

// D = A(16x4 f32) * B(4x16 f32) + C(16x16 f32), exact fp32 via matrix pipe.
static __device__ inline v8f wmma4(v2f a, v2f b, v8f c) {
  return __builtin_amdgcn_wmma_f32_16x16x4_f32(false, a, false, b, (short)0, c,
                                               false, false);
}

static __device__ inline float relu(float x) { return x > 0.f ? x : 0.f; }
static __device__ inline float sigm(float x) { return 1.f / (1.f + __expf(-x)); }

// --------------------------- async LDS staging -----------------------------
// gfx1250 ASYNCcnt-tracked global->LDS copies (CDNA5 ISA),
// guarded so a missing builtin falls back to plain load/store.
// Probe result (round 2 diagnostic): builtin exists, parameters are int*.
#if __has_builtin(__builtin_amdgcn_global_load_async_to_lds_b32) && \
    __has_builtin(__builtin_amdgcn_s_wait_asynccnt)
#define HAS_ASYNC_LDS 1
typedef __attribute__((address_space(1))) int as1_i32;
typedef __attribute__((address_space(3))) int as3_i32;
#else
#define HAS_ASYNC_LDS 0
#endif

static __device__ inline void stage_f32(float* dst_lds, const float* src_g) {
#if HAS_ASYNC_LDS
  __builtin_amdgcn_global_load_async_to_lds_b32(
      (as1_i32*)(int*)const_cast<float*>(src_g), (as3_i32*)(int*)dst_lds, 0, 0);
#else
  *dst_lds = *src_g;
#endif
}
static __device__ inline void stage_wait() {
#if HAS_ASYNC_LDS
  __builtin_amdgcn_s_wait_asynccnt(0);
#endif
}

// B pre-swizzle: element B[k][n] (row-major [K,N]) goes to LDS index
//   (((k>>2)*2 + (k&1))*N + n)*2 + ((k>>1)&1)
// so the v2f fragment {B[k0+hi][n], B[k0+2+hi][n]} is one aligned b64 load.
static __device__ inline int bswz(int k, int n, int N) {
  return ((((k >> 2) * 2 + (k & 1)) * N + n) << 1) + ((k >> 1) & 1);
}

// ---------------------------------------------------------------------------
// Kernel 1: conv1 (10x10x3 -> 32, y stride 1, x stride 2) + ReLU + 2x2 maxpool, fused.
// Grid: (98, 16)  block: 256.  One block = one (batch, pooled-row) strip:
// conv rows {2*py, 2*py+1} x 296 cols -> 592 pixels = 37 M-tiles.  K=300,N=32.
// ---------------------------------------------------------------------------
__global__ __launch_bounds__(256) void conv1_pool_kernel(
    const float* __restrict__ img, const float* __restrict__ w1,
    const float* __restrict__ b1, float* __restrict__ x1) {
  extern __shared__ float lds[];
  float* Bsw = lds;           // 300*32 = 9600 floats, swizzled
  float* Csh = lds + 9600;    // 592*32 = 18944 floats
  const int py = blockIdx.x, b = blockIdx.y;
  const int tid = threadIdx.x;

  for (int i = tid; i < 9600; i += 256)
    stage_f32(&Bsw[bswz(i >> 5, i & 31, 32)], &w1[i]);
  stage_wait();
  __syncthreads();

  const int wave = tid >> 5, lane = tid & 31;
  const int hi = lane >> 4, lo = lane & 15;
  const float* imgb = img + b * 720000;  // 400*600*3

  for (int tile = wave; tile < 37; tile += 8) {
    int m = tile * 16 + lo;
    int row1 = (m >= 296) ? 1 : 0;
    int oy = 2 * py + row1;
    int ox = m - 296 * row1;
    int pixBase = oy * 1800 + ox * 6;

    v8f acc0 = {}; v8f acc1 = {};
    for (int k0 = 0; k0 < 300; k0 += 4) {
      int ka = k0 + 2 * hi;          // even -> never crosses a 30-float row
      int ky = ka / 30;
      v2f a = *(const v2f*)&imgb[pixBase + ky * 1800 + (ka - ky * 30)];
      int q2 = (k0 >> 2) * 2 + hi;
      v2f bb0 = *(const v2f*)&Bsw[((q2 * 32 + lo) << 1)];
      v2f bb1 = *(const v2f*)&Bsw[((q2 * 32 + 16 + lo) << 1)];
      acc0 = wmma4(a, bb0, acc0);
      acc1 = wmma4(a, bb1, acc1);
    }
    float bias0 = b1[lo], bias1 = b1[16 + lo];
#pragma unroll
    for (int r = 0; r < 8; ++r) {
      int mm = tile * 16 + r + 8 * hi;  // D layout: VGPR r -> M = r + 8*hi
      Csh[mm * 32 + lo] = relu(acc0[r] + bias0);
      Csh[mm * 32 + 16 + lo] = relu(acc1[r] + bias1);
    }
  }
  __syncthreads();

  float* out = x1 + (size_t)(b * 98 + py) * 148 * 32;
  for (int i = tid; i < 4736; i += 256) {  // 148*32
    int pc = i >> 5, c = i & 31;
    int m0 = 2 * pc;
    float v = Csh[m0 * 32 + c];
    v = fmaxf(v, Csh[(m0 + 1) * 32 + c]);
    v = fmaxf(v, Csh[(296 + m0) * 32 + c]);
    v = fmaxf(v, Csh[(297 + m0) * 32 + c]);
    out[i] = v;
  }
}

// ---------------------------------------------------------------------------
// Kernel 2: conv2 (4x4x32 -> 64, stride 4) + ReLU + 2x2 maxpool, fused.
// Grid: (12, 16)  block: 256.  74 pixels -> 5 M-tiles.  K=512, N=64.
// ---------------------------------------------------------------------------
__global__ __launch_bounds__(256) void conv2_pool_kernel(
    const float* __restrict__ x1, const float* __restrict__ w2,
    const float* __restrict__ b2, float* __restrict__ x2) {
  extern __shared__ float lds[];
  float* Bsw = lds;            // 512*64 = 32768 floats, swizzled
  float* Csh = lds + 32768;    // 80*64  = 5120 floats
  const int py = blockIdx.x, b = blockIdx.y;
  const int tid = threadIdx.x;

  for (int i = tid; i < 32768; i += 256)
    stage_f32(&Bsw[bswz(i >> 6, i & 63, 64)], &w2[i]);
  stage_wait();
  __syncthreads();

  const int wave = tid >> 5, lane = tid & 31;
  const int hi = lane >> 4, lo = lane & 15;
  const float* inb = x1 + (size_t)b * 98 * 148 * 32;

  for (int tile = wave; tile < 5; tile += 8) {
    int m = tile * 16 + lo;
    int mc = m < 74 ? m : 73;
    int oy = 2 * py + mc / 37;
    int ox = mc % 37;
    int pixBase = 4 * oy * 4736 + 128 * ox;  // row stride 148*32

    v8f acc[4] = {v8f{}, v8f{}, v8f{}, v8f{}};
    for (int k0 = 0; k0 < 512; k0 += 4) {
      int ka = k0 + 2 * hi;  // even -> pair stays inside the 128-float run
      v2f a = *(const v2f*)&inb[pixBase + (ka >> 7) * 4736 + (ka & 127)];
      int q2 = (k0 >> 2) * 2 + hi;
#pragma unroll
      for (int nt = 0; nt < 4; ++nt) {
        v2f bb = *(const v2f*)&Bsw[((q2 * 64 + nt * 16 + lo) << 1)];
        acc[nt] = wmma4(a, bb, acc[nt]);
      }
    }
#pragma unroll
    for (int nt = 0; nt < 4; ++nt) {
      float bias = b2[nt * 16 + lo];
#pragma unroll
      for (int r = 0; r < 8; ++r) {
        int mm = tile * 16 + r + 8 * hi;
        if (mm < 74) Csh[mm * 64 + nt * 16 + lo] = relu(acc[nt][r] + bias);
      }
    }
  }
  __syncthreads();

  float* out = x2 + (size_t)(b * 12 + py) * 18 * 64;
  for (int i = tid; i < 1152; i += 256) {  // 18*64
    int pc = i >> 6, c = i & 63;
    int m0 = 2 * pc;
    float v = Csh[m0 * 64 + c];
    v = fmaxf(v, Csh[(m0 + 1) * 64 + c]);
    v = fmaxf(v, Csh[(37 + m0) * 64 + c]);
    v = fmaxf(v, Csh[(38 + m0) * 64 + c]);
    out[i] = v;
  }
}

// ---------------------------------------------------------------------------
// Kernel 3: conv3 (4x4x64 -> 128, stride 4) + ReLU -> x3 [16,1536].
// Grid: (8, 12) single-wave blocks.  M=192, N=128, K=1024.
// ---------------------------------------------------------------------------
__global__ __launch_bounds__(32) void conv3_kernel(
    const float* __restrict__ x2, const float* __restrict__ w3,
    const float* __restrict__ b3, float* __restrict__ x3) {
  const int nt = blockIdx.x, mt = blockIdx.y;
  const int lane = threadIdx.x & 31;
  const int hi = lane >> 4, lo = lane & 15;

  int m = mt * 16 + lo;  // m = b*12 + oy*4 + ox
  int b = m / 12, pix = m % 12;
  int oy = pix >> 2, ox = pix & 3;
  int pixBase = b * 13824 + 4 * oy * 1152 + 256 * ox;  // row stride 18*64

  v8f acc = {};
  for (int k0 = 0; k0 < 1024; k0 += 4) {
    int ka = k0 + 2 * hi;  // even -> pair inside the 256-float run
    v2f a = *(const v2f*)&x2[pixBase + (ka >> 8) * 1152 + (ka & 255)];
    int krow = k0 + hi;
    v2f bb;
    bb.x = w3[krow * 128 + nt * 16 + lo];
    bb.y = w3[(krow + 2) * 128 + nt * 16 + lo];
    acc = wmma4(a, bb, acc);
  }
  int n = nt * 16 + lo;
  float bias = b3[n];
#pragma unroll
  for (int r = 0; r < 8; ++r) {
    int mo = mt * 16 + r + 8 * hi;
    x3[(mo / 12) * 1536 + (mo % 12) * 128 + n] = relu(acc[r] + bias);
  }
}

// ---------------------------------------------------------------------------
// Kernel 4: dense [16,1536] x [1536,48000] + bias + ReLU -> d_out (x part).
// Memory-bound on wd (295 MB, read exactly once); one 16-wide N tile per
// wave, fp32 WMMA, global_prefetch_b8 streaming ahead.
// ---------------------------------------------------------------------------
__global__ __launch_bounds__(256) void dense_kernel(
    const float* __restrict__ x3, const float* __restrict__ wd,
    const float* __restrict__ bd, float* __restrict__ y) {
  extern __shared__ float Ash[];  // 16*1536 floats
  const int tid = threadIdx.x;
  for (int i = tid; i < 24576; i += 256) stage_f32(&Ash[i], &x3[i]);
  stage_wait();
  __syncthreads();

  const int wave = tid >> 5, lane = tid & 31;
  const int hi = lane >> 4, lo = lane & 15;
  const int ntile = blockIdx.x * 8 + wave;  // 0..2999
  const int col = ntile * 16 + lo;

  v8f acc = {};
  for (int k0 = 0; k0 < 1536; k0 += 4) {
    int ka = k0 + 2 * hi;
    v2f a = *(const v2f*)&Ash[lo * 1536 + ka];
    int krow = k0 + hi;
    if (k0 + 64 < 1536)  // stream wd: global_prefetch_b8
      __builtin_prefetch(&wd[(k0 + 64 + hi) * 48000 + col], 0, 1);
    v2f bb;
    bb.x = wd[krow * 48000 + col];
    bb.y = wd[(krow + 2) * 48000 + col];
    acc = wmma4(a, bb, acc);
  }
  float bias = bd[col];
#pragma unroll
  for (int r = 0; r < 8; ++r) {
    int mo = r + 8 * hi;  // batch row
    y[(size_t)mo * 48000 + col] = relu(acc[r] + bias);
  }
}

// ---------------------------------------------------------------------------
// Kernel 5: predict_transform + filter_boxes, faithful to the reference's
// raw reshape of the stacked tensors.  j in [0, 76800).
// ---------------------------------------------------------------------------
__global__ __launch_bounds__(256) void transform_kernel(
    const float* __restrict__ p, float* __restrict__ bboxes,
    float* __restrict__ bscores, float* __restrict__ bclasses) {
  int j = blockIdx.x * blockDim.x + threadIdx.x;
  if (j >= 76800) return;

  float ctr[2], whv[2];
#pragma unroll
  for (int c = 0; c < 2; ++c) {
    int o = j * 2 + c;  // flat into [B,GY,GX,2,2]
    int s = o / 76800; int r = o % 76800;
    int bp = r / 4800; r %= 4800;
    int gyp = r / 120; r %= 120;
    int gxp = r >> 1; int cp = r & 1;
    const float* pv = p + ((bp * 40 + gyp) * 60 + gxp) * 20;
    ctr[c] = sigm(pv[(s ? 10 : 0) + cp]) + (cp == 0 ? (float)gxp : (float)gyp);
    float pw = pv[(s ? 12 : 2) + cp];
    int ch = (s ? 2 : 0) + cp;
    int h = (4 * gxp + ch) >= 120 ? 1 : 0;
    float anchor = (h == (ch & 1)) ? 1.0f : 0.5f;
    whv[c] = __expf(pw) * anchor;
  }

  int so = j / 38400; int ro = j % 38400;
  int bo = ro / 2400; ro %= 2400;
  int gyo = ro / 60; int gxo = ro % 60;
  float obj = sigm(p[((bo * 40 + gyo) * 60 + gxo) * 20 + (so ? 14 : 4)]);

  float best = -1e30f; int besti = 0;
#pragma unroll
  for (int c5 = 0; c5 < 5; ++c5) {
    int o = j * 5 + c5;  // flat into [B,GY,GX,2,5]
    int s = o / 192000; int r = o % 192000;
    int bp = r / 12000; r %= 12000;
    int gyp = r / 300; r %= 300;
    int gxp = r / 5; int cp = r % 5;
    float sc =
        obj * sigm(p[((bp * 40 + gyp) * 60 + gxp) * 20 + (s ? 15 : 5) + cp]);
    if (sc > best) { best = sc; besti = c5; }
  }

  float minx = ctr[0] - 0.5f * whv[0], miny = ctr[1] - 0.5f * whv[1];
  float maxx = ctr[0] + 0.5f * whv[0], maxy = ctr[1] + 0.5f * whv[1];
  bboxes[j * 4 + 0] = miny * 400.f;
  bboxes[j * 4 + 1] = minx * 600.f;
  bboxes[j * 4 + 2] = maxy * 400.f;
  bboxes[j * 4 + 3] = maxx * 600.f;
  bscores[j] = (best >= 0.3f) ? best : -1e9f;
  bclasses[j] = (float)besti;
}

// ---------------------------------------------------------------------------
// Kernel 6: padded NMS (20 rounds of argmax + IoU suppression), 1 block.
// ---------------------------------------------------------------------------
__global__ __launch_bounds__(1024) void nms_kernel(
    const float* __restrict__ bboxes, float* __restrict__ bscores,
    const float* __restrict__ bclasses, float* __restrict__ out) {
  __shared__ float sval[1024];
  __shared__ int sidx[1024];
  __shared__ float cbox[5];
  __shared__ int cidx;
  __shared__ float selS[20];
  __shared__ int selI[20];
  const int tid = threadIdx.x;

  for (int t = 0; t < 20; ++t) {
    float bv = -3e38f; int bi = 0x7fffffff;
    for (int j = tid; j < 76800; j += 1024) {
      float v = bscores[j];
      if (v > bv || (v == bv && j < bi)) { bv = v; bi = j; }
    }
    sval[tid] = bv; sidx[tid] = bi;
    __syncthreads();
    for (int off = 512; off > 0; off >>= 1) {
      if (tid < off) {
        float v = sval[tid + off]; int i2 = sidx[tid + off];
        if (v > sval[tid] || (v == sval[tid] && i2 < sidx[tid])) {
          sval[tid] = v; sidx[tid] = i2;
        }
      }
      __syncthreads();
    }
    if (tid == 0) {
      int i = sidx[0];
      selS[t] = sval[0]; selI[t] = i; cidx = i;
      float y1 = bboxes[i * 4], x1 = bboxes[i * 4 + 1];
      float y2 = bboxes[i * 4 + 2], x2 = bboxes[i * 4 + 3];
      cbox[0] = y1; cbox[1] = x1; cbox[2] = y2; cbox[3] = x2;
      cbox[4] = fmaxf(y2 - y1, 0.f) * fmaxf(x2 - x1, 0.f);
    }
    __syncthreads();
    float y1 = cbox[0], x1 = cbox[1], y2 = cbox[2], x2 = cbox[3], ai = cbox[4];
    int isel = cidx;
    for (int j = tid; j < 76800; j += 1024) {
      float by1 = bboxes[j * 4], bx1 = bboxes[j * 4 + 1];
      float by2 = bboxes[j * 4 + 2], bx2 = bboxes[j * 4 + 3];
      float yy1 = fmaxf(y1, by1), xx1 = fmaxf(x1, bx1);
      float yy2 = fminf(y2, by2), xx2 = fminf(x2, bx2);
      float inter = fmaxf(yy2 - yy1, 0.f) * fmaxf(xx2 - xx1, 0.f);
      float aj = fmaxf(by2 - by1, 0.f) * fmaxf(bx2 - bx1, 0.f);
      float iou = inter / (ai + aj - inter + 1e-9f);
      if (iou > 0.4f || j == isel) bscores[j] = -1e9f;
    }
    __syncthreads();
  }

  if (tid < 20) {
    float s = selS[tid]; int i = selI[tid];
    bool valid = s > -5e8f;  // NEG/2
#pragma unroll
    for (int c = 0; c < 4; ++c)
      out[768000 + tid * 4 + c] = valid ? bboxes[i * 4 + c] : 0.f;
    out[768080 + tid] = valid ? s : 0.f;
    out[768100 + tid] = valid ? (float)bclasses[i] : 0.f;
  }
}

// ---------------------------------------------------------------------------
extern "C" void kernel_launch(void* const* d_in, const int* in_sizes, int n_in,
                              void* d_out, int out_size, void* d_ws,
                              size_t ws_size, hipStream_t stream) {
  (void)in_sizes; (void)n_in; (void)out_size; (void)ws_size;
  const float* img = (const float*)d_in[0];
  const float* w1 = (const float*)d_in[1];
  const float* b1 = (const float*)d_in[2];
  const float* w2 = (const float*)d_in[3];
  const float* b2 = (const float*)d_in[4];
  const float* w3 = (const float*)d_in[5];
  const float* b3 = (const float*)d_in[6];
  const float* wd = (const float*)d_in[7];
  const float* bd = (const float*)d_in[8];
  float* out = (float*)d_out;

  float* ws = (float*)d_ws;
  float* x1 = ws;                    // 16*98*148*32 = 7,426,048
  float* x2 = x1 + 7426048;          // 16*12*18*64  =   221,184
  float* x3 = x2 + 221184;           // 16*1536      =    24,576
  float* bboxes = x3 + 24576;        // 76800*4      =   307,200
  float* bscores = bboxes + 307200;  // 76,800
  float* bclasses = bscores + 76800; // 76,800

  conv1_pool_kernel<<<dim3(98, 16), 256, (9600 + 18944) * 4, stream>>>(
      img, w1, b1, x1);
  conv2_pool_kernel<<<dim3(12, 16), 256, (32768 + 5120) * 4, stream>>>(
      x1, w2, b2, x2);
  conv3_kernel<<<dim3(8, 12), 32, 0, stream>>>(x2, w3, b3, x3);
  dense_kernel<<<375, 256, 24576 * 4, stream>>>(x3, wd, bd, out);
  transform_kernel<<<300, 256, 0, stream>>>(out, bboxes, bscores, bclasses);
  nms_kernel<<<1, 1024, 0, stream>>>(bboxes, bscores, bclasses, out);
}